// splineconv_66228395705232
// MI455X (gfx1250) — compile-verified
//
#include <hip/hip_runtime.h>
#include <math.h>

typedef __attribute__((ext_vector_type(2))) float v2f;
typedef __attribute__((ext_vector_type(8))) float v8f;

#define N_NODES 50000
#define N_EDGES 800000

// ---------------- pack weights, transposed: Bt[col][k] = W[k][col] ----------------
// Bt1: 384x128  (cols 0-127 = W1[0], 128-255 = W1[1], 256-383 = root1)
// Bt2:  96x128  (cols 0-31  = W2[0], 32-63   = W2[1], 64-95   = root2)
__global__ __launch_bounds__(256) void pack_weights(
    const float* __restrict__ W1, const float* __restrict__ root1,
    const float* __restrict__ W2, const float* __restrict__ root2,
    float* __restrict__ Bt1, float* __restrict__ Bt2) {
  int i = blockIdx.x * 256 + threadIdx.x;
  const int tot1 = 384 * 128, tot2 = 96 * 128;
  if (i < tot1) {
    int col = i >> 7, k = i & 127;
    float v;
    if (col < 128)      v = W1[k * 128 + col];
    else if (col < 256) v = W1[128 * 128 + k * 128 + (col - 128)];
    else                v = root1[k * 128 + (col - 256)];
    Bt1[col * 128 + k] = v;
  } else if (i < tot1 + tot2) {
    int j = i - tot1;
    int col = j >> 7, k = j & 127;
    float v;
    if (col < 32)      v = W2[k * 32 + col];
    else if (col < 64) v = W2[128 * 32 + k * 32 + (col - 32)];
    else               v = root2[k * 32 + (col - 64)];
    Bt2[col * 128 + k] = v;
  }
}

__global__ __launch_bounds__(256) void zero_f32(float* __restrict__ p, int n) {
  int i = blockIdx.x * 256 + threadIdx.x;
  if (i < n) p[i] = 0.0f;
}

// ---------------- fp32 WMMA GEMM: C[M x N] = A[M x 128] * Bt[N x 128]^T ----------------
// M multiple of 16 (50000 = 3125*16). One block = one 16-row M tile, 8 waves sweep N tiles.
template <int NTILES>
__global__ __launch_bounds__(256) void gemm_wmma_f32(
    const float* __restrict__ A, const float* __restrict__ Bt, float* __restrict__ C) {
  __shared__ float lds[16 * 132];  // padded stride 132 dwords -> conflict-free b64 reads
  const int tid  = threadIdx.x;
  const int lane = tid & 31;
  const int wave = tid >> 5;
  const int mbase = blockIdx.x * 16;

  // Stage 16x128 A tile into LDS with coalesced float4 loads.
  const float4* A4 = (const float4*)(A + (size_t)mbase * 128);
#pragma unroll
  for (int it = 0; it < 2; ++it) {
    int i  = tid + it * 256;       // 0..511 float4s
    int r  = i >> 5;               // row 0..15
    int c4 = i & 31;               // float4 col 0..31
    float4 v = A4[r * 32 + c4];
    *(float4*)&lds[r * 132 + c4 * 4] = v;
  }
  __syncthreads();

  const int lhalf = lane & 15;
  const int khi   = (lane < 16) ? 0 : 2;  // ISA A-layout: high lane half holds K+2,K+3
  const float* arow = &lds[lhalf * 132 + khi];

  for (int nt = wave; nt < NTILES; nt += 8) {
    const float* brow = Bt + (size_t)(nt * 16 + lhalf) * 128 + khi;
    v8f acc = {};
#pragma unroll
    for (int k0 = 0; k0 < 128; k0 += 4) {
      v2f a = *(const v2f*)(arow + k0);
      v2f b = *(const v2f*)(brow + k0);
      acc = __builtin_amdgcn_wmma_f32_16x16x4_f32(false, a, false, b, (short)0, acc,
                                                  false, false);
    }
    // C/D layout: VGPR i -> M = i (lanes 0-15) or i+8 (lanes 16-31), N = lane&15
    const int rbase = mbase + ((lane < 16) ? 0 : 8);
    const int cg = nt * 16 + lhalf;
#pragma unroll
    for (int i = 0; i < 8; ++i)
      C[(size_t)(rbase + i) * (NTILES * 16) + cg] = acc[i];
  }
}

// ---------------- layer 1 edge phase: one wave per edge, 128 feats (4/lane) ----------------
__global__ __launch_bounds__(256) void edge_layer1(
    const int* __restrict__ ei, const float* __restrict__ u,
    const float* __restrict__ XW,  // N_NODES x 384 (L2-resident: 76.8MB < 192MB L2)
    float* __restrict__ agg,       // N_NODES x 128
    float* __restrict__ cnt) {
  int wave = threadIdx.x >> 5, lane = threadIdx.x & 31;
  int e = blockIdx.x * 8 + wave;  // grid = 100000 -> exactly 800000 edges
  int src = ei[e];
  int dst = ei[N_EDGES + e];
  float uu = u[e], w0 = 1.0f - uu;
  const float4* p = (const float4*)(XW + (size_t)src * 384);
  float4 x0 = p[lane];        // k=0 slice, feats 4*lane..
  float4 x1 = p[32 + lane];   // k=1 slice (+128 floats)
  float4 m;
  m.x = w0 * x0.x + uu * x1.x;
  m.y = w0 * x0.y + uu * x1.y;
  m.z = w0 * x0.z + uu * x1.z;
  m.w = w0 * x0.w + uu * x1.w;
  float* ap = agg + (size_t)dst * 128 + lane * 4;
  atomicAdd(ap + 0, m.x);
  atomicAdd(ap + 1, m.y);
  atomicAdd(ap + 2, m.z);
  atomicAdd(ap + 3, m.w);
  if (lane == 0) atomicAdd(cnt + dst, 1.0f);
}

// ---------------- layer 1 finish: normalize + root + bias + relu ----------------
__global__ __launch_bounds__(256) void finish1(
    const float* __restrict__ agg, const float* __restrict__ XW,
    const float* __restrict__ b1, const float* __restrict__ cnt,
    float* __restrict__ h) {
  int i = blockIdx.x * 256 + threadIdx.x;  // 6,400,000 exactly
  int n = i >> 7, f = i & 127;
  float c = cnt[n];
  c = (c < 1.0f) ? 1.0f : c;
  float v = agg[i] / c + XW[(size_t)n * 384 + 256 + f] + b1[f];
  h[i] = fmaxf(v, 0.0f);
}

// ---------------- layer 2 edge phase: one wave per edge, 32 feats (1/lane) ----------------
__global__ __launch_bounds__(256) void edge_layer2(
    const int* __restrict__ ei, const float* __restrict__ u,
    const float* __restrict__ XW2,  // N_NODES x 96
    float* __restrict__ agg2) {     // N_NODES x 32
  int wave = threadIdx.x >> 5, lane = threadIdx.x & 31;
  int e = blockIdx.x * 8 + wave;
  int src = ei[e];
  int dst = ei[N_EDGES + e];
  float uu = u[e];
  const float* p = XW2 + (size_t)src * 96;
  float x0 = p[lane];
  float x1 = p[32 + lane];
  atomicAdd(agg2 + (size_t)dst * 32 + lane, (1.0f - uu) * x0 + uu * x1);
}

// ---------------- layer 2 finish: normalize + root + bias + relu + log_softmax ----------------
__global__ __launch_bounds__(256) void finish2(
    const float* __restrict__ agg2, const float* __restrict__ XW2,
    const float* __restrict__ b2, const float* __restrict__ cnt,
    float* __restrict__ out) {
  int wave = threadIdx.x >> 5, lane = threadIdx.x & 31;  // lane == class
  int n = blockIdx.x * 8 + wave;  // grid = 6250 -> exactly 50000 nodes
  float c = cnt[n];
  c = (c < 1.0f) ? 1.0f : c;
  float v = agg2[(size_t)n * 32 + lane] / c + XW2[(size_t)n * 96 + 64 + lane] + b2[lane];
  v = fmaxf(v, 0.0f);
  // wave32 reductions for log_softmax over the 32 classes
  float m = v;
#pragma unroll
  for (int o = 16; o > 0; o >>= 1) m = fmaxf(m, __shfl_xor(m, o, 32));
  float s = expf(v - m);
#pragma unroll
  for (int o = 16; o > 0; o >>= 1) s += __shfl_xor(s, o, 32);
  out[(size_t)n * 32 + lane] = v - m - logf(s);
}

extern "C" void kernel_launch(void* const* d_in, const int* in_sizes, int n_in,
                              void* d_out, int out_size, void* d_ws, size_t ws_size,
                              hipStream_t stream) {
  const float* x     = (const float*)d_in[0];
  const int*   ei    = (const int*)d_in[1];
  const float* ea    = (const float*)d_in[2];
  const float* W1    = (const float*)d_in[3];
  const float* root1 = (const float*)d_in[4];
  const float* b1    = (const float*)d_in[5];
  const float* W2    = (const float*)d_in[6];
  const float* root2 = (const float*)d_in[7];
  const float* b2    = (const float*)d_in[8];
  float* out = (float*)d_out;

  // Workspace layout (floats). XW2/agg2 reuse the XW1 region after finish1
  // (stream order serializes the hazard). Peak = 32,111,440 floats (~128.4 MB).
  float* ws   = (float*)d_ws;
  float* XW1  = ws;               // 19,200,000  (50000 x 384)
  float* XW2  = ws;               //  4,800,000  (50000 x 96)   [reuse]
  float* agg2 = ws + 4800000;     //  1,600,000  (50000 x 32)   [reuse]
  float* agg1 = ws + 19200000;    //  6,400,000  (50000 x 128)
  float* h    = ws + 25600000;    //  6,400,000  (50000 x 128)
  float* cnt  = ws + 32000000;    //     50,000
  float* Bt1  = ws + 32050000;    //     49,152  (384 x 128)
  float* Bt2  = ws + 32099152;    //     12,288  (96 x 128)
  if (ws_size < (size_t)32111440 * sizeof(float)) return;

  pack_weights<<<240, 256, 0, stream>>>(W1, root1, W2, root2, Bt1, Bt2);
  zero_f32<<<(6400000 + 255) / 256, 256, 0, stream>>>(agg1, 6400000);
  zero_f32<<<(50000 + 255) / 256, 256, 0, stream>>>(cnt, 50000);

  gemm_wmma_f32<24><<<3125, 256, 0, stream>>>(x, Bt1, XW1);       // 50000 x 384
  edge_layer1<<<100000, 256, 0, stream>>>(ei, ea, XW1, agg1, cnt);
  finish1<<<25000, 256, 0, stream>>>(agg1, XW1, b1, cnt, h);

  gemm_wmma_f32<6><<<3125, 256, 0, stream>>>(h, Bt2, XW2);        // 50000 x 96
  zero_f32<<<(1600000 + 255) / 256, 256, 0, stream>>>(agg2, 1600000);
  edge_layer2<<<100000, 256, 0, stream>>>(ei, ea, XW2, agg2);
  finish2<<<6250, 256, 0, stream>>>(agg2, XW2, b2, cnt, out);
}